// GINLayer_73478300500080
// MI455X (gfx1250) — compile-verified
//
#include <hip/hip_runtime.h>
#include <hip/hip_bf16.h>

typedef float v2f __attribute__((ext_vector_type(2)));
typedef float v8f __attribute__((ext_vector_type(8)));

#define DFEAT 128
#define WT_STRIDE 129   // padded LDS stride for transposed weights (kills stride-128 bank conflicts)
#define HIN_STRIDE 132  // padded LDS stride for the 16-row input tile
#define BN_SLICES 32    // spread BN-partial atomics over 32 slices per column

// ---------------------------------------------------------------------------
// Kernel 1: edge scatter-add.  One wave32 per edge: 32 lanes x float4 = the
// full 128-float source row; 4 global_atomic_add_f32 per lane (no return ->
// STOREcnt fire-and-forget).  src/dst indices are wave-uniform -> scalar loads.
// ---------------------------------------------------------------------------
__global__ void __launch_bounds__(256)
gin_scatter_kernel(const float* __restrict__ x,
                   const int* __restrict__ src,
                   const int* __restrict__ dst,
                   float* __restrict__ agg, int E)
{
    int t = blockIdx.x * 256 + threadIdx.x;
    int e = t >> 5;
    if (e >= E) return;
    int lane = t & 31;
    int s = src[e];
    int d = dst[e];
    const float4 v = ((const float4*)(x + (size_t)s * DFEAT))[lane];
    float* ad = agg + (size_t)d * DFEAT + lane * 4;
    atomicAdd(ad + 0, v.x);
    atomicAdd(ad + 1, v.y);
    atomicAdd(ad + 2, v.z);
    atomicAdd(ad + 3, v.w);
}

// ---------------------------------------------------------------------------
// Kernels 2 & 4: fused GEMM via V_WMMA_F32_16X16X4_F32.
// MODE 0:  hin = (1+eps)*x + agg;           out = hin @ W1^T + b1 ; BN partials
// MODE 1:  hin = relu(scale*h1 + shift);    out = hin @ W2^T + b2
// Block = 256 threads (8 wave32), one 16-row strip; wave w owns column tile
// n0 = 16*w.  K=128 chained through 32 WMMA steps of K=4.
// Dynamic LDS: Wt[128][129] (W transposed, [k][n]) + hin[16][132].
// ---------------------------------------------------------------------------
extern __shared__ float lds_buf[];

template <int MODE>
__global__ void __launch_bounds__(256)
gin_gemm_kernel(const float* __restrict__ Ax,      // x (MODE0) or h1 (MODE1)
                const float* __restrict__ Aagg,    // agg (MODE0)
                const float* __restrict__ epsp,    // eps scalar (MODE0)
                const float* __restrict__ scale,   // BN scale (MODE1)
                const float* __restrict__ shift,   // BN shift (MODE1)
                const float* __restrict__ W,       // [128][128] row-major [n][k]
                const float* __restrict__ bias,    // [128]
                float* __restrict__ out,           // h1 or final output
                float* __restrict__ bn_sum,        // [BN_SLICES][128] (MODE0)
                float* __restrict__ bn_sumsq,      // [BN_SLICES][128] (MODE0)
                int Nrows)
{
    float* Wt  = lds_buf;                       // [k][n], stride WT_STRIDE
    float* hin = lds_buf + DFEAT * WT_STRIDE;   // [m][k], stride HIN_STRIDE

    const int tid  = threadIdx.x;
    const int row0 = blockIdx.x * 16;

    // Stage W transposed: Wt[k][n] = W[n][k]  (coalesced global reads)
    for (int i = tid; i < DFEAT * DFEAT; i += 256) {
        int n = i >> 7;
        int k = i & 127;
        Wt[k * WT_STRIDE + n] = W[n * DFEAT + k];
    }

    // Build the 16x128 input tile with the fused elementwise op
    float e1 = 0.0f;
    if (MODE == 0) e1 = 1.0f + epsp[0];
    for (int i = tid; i < 16 * DFEAT; i += 256) {
        int m = i >> 7;
        int k = i & 127;
        int gr = row0 + m;
        float v = 0.0f;
        if (gr < Nrows) {
            size_t g = (size_t)gr * DFEAT + k;
            if (MODE == 0) {
                v = e1 * Ax[g] + Aagg[g];
            } else {
                v = fmaxf(scale[k] * Ax[g] + shift[k], 0.0f);
            }
        }
        hin[m * HIN_STRIDE + k] = v;
    }
    __syncthreads();

    const int wave  = tid >> 5;            // 0..7
    const int lane  = tid & 31;
    const int lm    = lane & 15;           // M for A, N for B/C/D
    const int khalf = (lane >> 4) << 1;    // 0 (lanes 0-15) or 2 (lanes 16-31)
    const int n0    = wave * 16;

    // Accumulator init = bias[n]  (C/D layout: N = lane%16 for every VGPR)
    v8f c;
    {
        float b = bias[n0 + lm];
        for (int r = 0; r < 8; ++r) c[r] = b;
    }

    const float* arow  = hin + lm * HIN_STRIDE + khalf;        // A: row M=lm
    const float* bbase = Wt + khalf * WT_STRIDE + n0 + lm;     // B: col N=n0+lm

    for (int k0 = 0; k0 < DFEAT; k0 += 4) {
        v2f a, b;
        a.x = arow[k0];                       // A[M][k0+khalf]
        a.y = arow[k0 + 1];                   // A[M][k0+khalf+1]
        b.x = bbase[k0 * WT_STRIDE];          // B[k0+khalf][N]
        b.y = bbase[(k0 + 1) * WT_STRIDE];    // B[k0+khalf+1][N]
        c = __builtin_amdgcn_wmma_f32_16x16x4_f32(
                /*neg_a=*/false, a, /*neg_b=*/false, b,
                /*c_mod=*/(short)0, c, /*reuse_a=*/false, /*reuse_b=*/false);
    }

    // Write the 16x16 tile; accumulate BN partials (MODE 0)
    const int mbase = (lane >> 4) << 3;       // 0 or 8
    float s = 0.0f, q = 0.0f;
    for (int r = 0; r < 8; ++r) {
        int gm = row0 + mbase + r;
        if (gm < Nrows) {
            float v = c[r];
            out[(size_t)gm * DFEAT + n0 + lm] = v;
            if (MODE == 0) { s += v; q += v * v; }
        }
    }
    if (MODE == 0) {
        s += __shfl_xor(s, 16, 32);           // combine M=0..7 with M=8..15
        q += __shfl_xor(q, 16, 32);
        if (lane < 16) {
            int slice = blockIdx.x & (BN_SLICES - 1);
            atomicAdd(&bn_sum  [slice * DFEAT + n0 + lm], s);
            atomicAdd(&bn_sumsq[slice * DFEAT + n0 + lm], q);
        }
    }
}

// ---------------------------------------------------------------------------
// Kernel 3: fold BN partial slices into per-column scale/shift.
// ---------------------------------------------------------------------------
__global__ void __launch_bounds__(128)
gin_bnstats_kernel(const float* __restrict__ bn_sum,
                   const float* __restrict__ bn_sumsq,
                   const float* __restrict__ gamma,
                   const float* __restrict__ beta,
                   float* __restrict__ scale,
                   float* __restrict__ shift,
                   float invN)
{
    int n = threadIdx.x;   // 128 threads
    float s = 0.0f, q = 0.0f;
    for (int i = 0; i < BN_SLICES; ++i) {
        s += bn_sum[i * DFEAT + n];
        q += bn_sumsq[i * DFEAT + n];
    }
    float mean = s * invN;
    float var  = q * invN - mean * mean;
    float inv  = rsqrtf(var + 1e-5f);
    float sc   = gamma[n] * inv;
    scale[n] = sc;
    shift[n] = beta[n] - mean * sc;
}

// ---------------------------------------------------------------------------
extern "C" void kernel_launch(void* const* d_in, const int* in_sizes, int n_in,
                              void* d_out, int out_size, void* d_ws, size_t ws_size,
                              hipStream_t stream) {
    const float* x     = (const float*)d_in[0];
    const int*   ei    = (const int*)  d_in[1];   // [2][E]: row0=src, row1=dst
    const float* W1    = (const float*)d_in[2];
    const float* b1    = (const float*)d_in[3];
    const float* gamma = (const float*)d_in[4];
    const float* beta  = (const float*)d_in[5];
    const float* W2    = (const float*)d_in[6];
    const float* b2    = (const float*)d_in[7];
    const float* eps   = (const float*)d_in[8];
    float* outp = (float*)d_out;

    const int E = in_sizes[1] / 2;
    const int N = in_sizes[0] / DFEAT;

    // Workspace layout
    char*  ws       = (char*)d_ws;
    size_t nd_bytes = (size_t)N * DFEAT * sizeof(float);
    float* agg      = (float*)ws;                          // [N][128]
    float* h1       = (float*)(ws + nd_bytes);             // [N][128]
    float* stats    = (float*)(ws + 2 * nd_bytes);         // BN scratch
    float* bn_sum   = stats;                               // [32][128]
    float* bn_sumsq = stats + BN_SLICES * DFEAT;           // [32][128]
    float* scale    = stats + 2 * BN_SLICES * DFEAT;       // [128]
    float* shift    = scale + DFEAT;                       // [128]

    // Self-initialize scratch every call (graph-capture safe)
    hipMemsetAsync(agg, 0, nd_bytes, stream);
    hipMemsetAsync(stats, 0, (2 * BN_SLICES * DFEAT + 2 * DFEAT) * sizeof(float), stream);

    // 1) scatter-add aggregation (one wave per edge)
    {
        long long threads = (long long)E * 32;
        int blocks = (int)((threads + 255) / 256);
        gin_scatter_kernel<<<blocks, 256, 0, stream>>>(x, ei, ei + E, agg, E);
    }

    const size_t ldsBytes = (size_t)(DFEAT * WT_STRIDE + 16 * HIN_STRIDE) * sizeof(float);
    const int gemmBlocks = (N + 15) / 16;

    // 2) h1 = ((1+eps)x + agg) @ W1^T + b1, plus BN partial sums
    gin_gemm_kernel<0><<<gemmBlocks, 256, ldsBytes, stream>>>(
        x, agg, eps, nullptr, nullptr, W1, b1, h1, bn_sum, bn_sumsq, N);

    // 3) BN stats -> per-column scale/shift
    gin_bnstats_kernel<<<1, 128, 0, stream>>>(
        bn_sum, bn_sumsq, gamma, beta, scale, shift, 1.0f / (float)N);

    // 4) out = relu(scale*h1 + shift) @ W2^T + b2
    gin_gemm_kernel<1><<<gemmBlocks, 256, ldsBytes, stream>>>(
        h1, nullptr, nullptr, scale, shift, W2, b2, outp, nullptr, nullptr, N);
}